// MagPotTorchScript_24472723652712
// MI455X (gfx1250) — compile-verified
//
#include <hip/hip_runtime.h>

#define KNBR 32
#define EMBED 16
#define PI_F 3.14159265358979f
#define RCUT 5.0f

typedef float v2f __attribute__((ext_vector_type(2)));
typedef float v8f __attribute__((ext_vector_type(8)));

// full-wave butterfly sum (only used for the final scalar energy)
__device__ __forceinline__ float wred32(float x) {
#pragma unroll
  for (int m = 16; m >= 1; m >>= 1) x += __shfl_xor(x, m, 32);
  return x;
}

// fused normalize + weight: ((v - shift[k]) * rcp(scale[k])) * w[EMBED + k]
__device__ __forceinline__ float nrm(float v, int k, const float* __restrict__ sh,
                                     const float* __restrict__ sc,
                                     const float* __restrict__ w) {
  return (v - sh[k]) * __builtin_amdgcn_rcpf(sc[k]) * w[EMBED + k];
}

#define LDS_PER_WAVE 1792   // floats; RED 256 | XS 768 | YS 768 (PH/FC phase-share XS/YS)

__global__ __launch_bounds__(256, 1)
void magpot_fused(const float* __restrict__ pos, const int* __restrict__ spec,
                  const float* __restrict__ mag, const int* __restrict__ eidx,
                  const float* __restrict__ shifts, const float* __restrict__ coeff,
                  const float* __restrict__ semb, const float* __restrict__ ss,
                  const float* __restrict__ ssc, const float* __restrict__ ms,
                  const float* __restrict__ msc, const float* __restrict__ esh,
                  const float* __restrict__ wst, const float* __restrict__ bst,
                  const float* __restrict__ wmg, const float* __restrict__ bmg,
                  float* __restrict__ out, int N) {
  const int wave = threadIdx.x >> 5;
  const int lane = threadIdx.x & 31;
  const int atom = blockIdx.x * 8 + wave;
  const int ia = (atom < N) ? atom : (N - 1);   // clamp: keep waves uniform for barriers
  const int E = N * KNBR;
  const int e = ia * KNBR + lane;               // one lane per edge

  __builtin_prefetch(coeff, 0, 3);              // hot tables -> global_prefetch
  __builtin_prefetch(wmg, 0, 3);
  __builtin_prefetch(wst, 0, 3);

  __shared__ float lds[8][LDS_PER_WAVE];
  float* const Lw  = &lds[wave][0];
  float* const RED = Lw;            // 16x16 segment-sum results, [row*16+col]
  float* const PH  = Lw + 256;      // stage-1 A staging: 32 lanes x 16 rows
  float* const FC  = Lw + 768;      // stage-1 B staging: 32 lanes x 16 factors
  float* const XS  = Lw + 256;      // X products -> (in-place) prefix sums S; [b*24 + d*8 + n]
  float* const YS  = Lw + 1024;     // DMI B staging: [b*24 + d*8 + m]

#define R_DR(n)  RED[(n) * 16 + 0]
#define R_DNB(n) RED[(8 + (n)) * 16 + 0]
#define R_P(n,d) RED[(n) * 16 + 1 + (d)]
#define R_M(n,d) RED[(n) * 16 + 4 + (d)]
#define R_W(n,d) RED[(n) * 16 + 7 + (d)]
#define R_Q(n,c) RED[(n) * 16 + 10 + (c)]

  // ---- central atom ----
  const float pix = pos[3 * ia + 0], piy = pos[3 * ia + 1], piz = pos[3 * ia + 2];
  const int spi = spec[ia];
  const float mix = mag[3 * ia + 0], miy = mag[3 * ia + 1], miz = mag[3 * ia + 2];
  const float u = mix * mix + miy * miy + miz * miz;
  const float mnrm = sqrtf(u);
  const float minv = (mnrm > 1e-8f) ? __builtin_amdgcn_rcpf(fmaxf(mnrm, 1e-8f)) : 0.0f;
  const float mdx = mix * minv, mdy = miy * minv, mdz = miz * minv;

  // ---- per-edge geometry ----
  const int j = eidx[E + e];
  float rr[3];
  rr[0] = pos[3 * j + 0] - pix + shifts[3 * e + 0];
  rr[1] = pos[3 * j + 1] - piy + shifts[3 * e + 1];
  rr[2] = pos[3 * j + 2] - piz + shifts[3 * e + 2];
  const float dist = sqrtf(rr[0] * rr[0] + rr[1] * rr[1] + rr[2] * rr[2]);
  const float dinv = __builtin_amdgcn_rcpf(fmaxf(dist, 1e-8f));
  const float hh[3] = { rr[0] * dinv, rr[1] * dinv, rr[2] * dinv };

  // ---- Chebyshev radial basis ----
  const float fcut = (dist < RCUT) ? (0.5f * __cosf(PI_F * dist * (1.0f / RCUT)) + 0.5f) : 0.0f;
  const float xx = 2.0f * dist * dist * (1.0f / (RCUT * RCUT)) - 1.0f;
  float T[8];
  T[0] = 1.0f; T[1] = xx;
#pragma unroll
  for (int n = 2; n < 8; ++n) T[n] = 2.0f * xx * T[n - 1] - T[n - 2];
  float fk[8];
#pragma unroll
  for (int k = 0; k < 8; ++k) fk[k] = 0.5f * (T[k] + 1.0f) * fcut;

  const int spj = spec[j];
  const float* C = coeff + (spi * 3 + spj) * 64;
  float phi[8];
#pragma unroll
  for (int n = 0; n < 8; ++n) {
    float s = 0.0f;
#pragma unroll
    for (int k = 0; k < 8; ++k) s = fmaf(C[n * 8 + k], fk[k], s);
    phi[n] = s;
  }

  // ---- neighbor magnetics ----
  const float mjx = mag[3 * j + 0], mjy = mag[3 * j + 1], mjz = mag[3 * j + 2];
  const float uj = mjx * mjx + mjy * mjy + mjz * mjz;
  const float mjn = sqrtf(uj);
  const float mjinv = (mjn > 1e-8f) ? __builtin_amdgcn_rcpf(fmaxf(mjn, 1e-8f)) : 0.0f;
  const float dj[3] = { mjx * mjinv, mjy * mjinv, mjz * mjinv };

  // =====================================================================
  // Stage 1: ALL segment sums as one K=32 GEMM on the matrix pipe.
  //   A (16x32): rows 0-7 = phi[n], rows 8-15 = phi[n]*u_j  (K = lane = edge)
  //   B (32x16): cols = [1, rhat(3), mdir_j(3), uj*mdir_j(3), traceless(6)]
  //   D (16x16): dr | P | M | W | Qu  (+ dnb in rows 8-15 col 0)
  // =====================================================================
#pragma unroll
  for (int n = 0; n < 8; ++n) {
    PH[lane * 16 + n]     = phi[n];
    PH[lane * 16 + 8 + n] = phi[n] * uj;
  }
  {
    float* F = FC + lane * 16;
    F[0] = 1.0f;
    F[1] = hh[0];  F[2] = hh[1];  F[3] = hh[2];
    F[4] = dj[0];  F[5] = dj[1];  F[6] = dj[2];
    F[7] = uj * dj[0]; F[8] = uj * dj[1]; F[9] = uj * dj[2];
    F[10] = hh[0] * hh[0] - (1.0f / 3.0f);
    F[11] = hh[1] * hh[1] - (1.0f / 3.0f);
    F[12] = hh[2] * hh[2] - (1.0f / 3.0f);
    F[13] = hh[0] * hh[1];
    F[14] = hh[0] * hh[2];
    F[15] = hh[1] * hh[2];
  }
  __syncthreads();

  const bool lo = (lane < 16);
  const int row16 = lane & 15;
  const int r8 = lane & 7;
  const int hi2 = (lane >> 4) << 1;   // 0 (lanes 0-15) or 2 (lanes 16-31): K sub-slot

  v8f acc = {};
#pragma unroll
  for (int s = 0; s < 8; ++s) {
    const int k0 = (s * 4 + hi2) * 16 + row16;
    const int k1 = k0 + 16;
    v2f a, b;
    a.x = PH[k0]; a.y = PH[k1];
    b.x = FC[k0]; b.y = FC[k1];
    acc = __builtin_amdgcn_wmma_f32_16x16x4_f32(false, a, false, b, (short)0, acc, false, false);
  }
  {
    const int rbase = lo ? 0 : 8;     // D: lanes 0-15 rows 0-7, lanes 16-31 rows 8-15
#pragma unroll
    for (int v = 0; v < 8; ++v) RED[(rbase + v) * 16 + row16] = acc[v];
  }
  __syncthreads();

  // =====================================================================
  // Stage 2: Gram matrices on WMMA.  PP = P.P^T (K=3),
  //          QQ = Qf.Qf^T (K=9 via symmetry-weighted K=6), SAE = Qm.M^T (K=3)
  // LDS loads unconditional; only padded-K slots zeroed.
  // =====================================================================
  const float wq = lo ? 1.0f : 2.0f;                 // off-diag symmetry weight
  const float p01 = R_P(r8, hi2);                    // P[r8][0|2]
  const float p1  = R_P(r8, 1);
  const float q02 = R_Q(r8, hi2);                    // Qu[r8][0|2]
  const float q13 = R_Q(r8, 1 + hi2);                // Qu[r8][1|3]
  const float q4  = R_Q(r8, 4), q5 = R_Q(r8, 5);
  const float m02 = R_M(r8, hi2);                    // M[r8][0|2]
  const float m1  = R_M(r8, 1);
  const float qu0 = R_Q(r8, 0), qu1 = R_Q(r8, 1), qu2 = R_Q(r8, 2);
  const float qu3 = R_Q(r8, 3);

  v2f aP, aQ1, bQ1, aQ2, bQ2, aS2, bS2;
  aP.x  = p01;
  aP.y  = lo ? p1 : 0.0f;                            // K=3 pad
  aQ1.x = q02;  aQ1.y = q13;                         // K 0..3 all real
  bQ1.x = q02;  bQ1.y = q13 * wq;                    // w(xy)=2
  aQ2.x = lo ? q4 : 0.0f;                            // K=6,7 pad (A side zero)
  aQ2.y = lo ? q5 : 0.0f;
  bQ2.x = 2.0f * q4;  bQ2.y = 2.0f * q5;             // w(xz)=w(yz)=2
  {
    const float qm0 = qu0 * mdx + qu3 * mdy + q4 * mdz;   // Qm = Q . m_dir
    const float qm1 = qu3 * mdx + qu1 * mdy + q5 * mdz;
    const float qm2 = q4 * mdx + q5 * mdy + qu2 * mdz;
    aS2.x = lo ? qm0 : qm2;
    aS2.y = lo ? qm1 : 0.0f;                         // K=3 pad
  }
  bS2.x = m02;
  bS2.y = lo ? m1 : 0.0f;

  v8f ppd = {}; v8f qqd = {}; v8f saed = {};
  ppd  = __builtin_amdgcn_wmma_f32_16x16x4_f32(false, aP,  false, aP,  (short)0, ppd,  false, false);
  qqd  = __builtin_amdgcn_wmma_f32_16x16x4_f32(false, aQ1, false, bQ1, (short)0, qqd,  false, false);
  qqd  = __builtin_amdgcn_wmma_f32_16x16x4_f32(false, aQ2, false, bQ2, (short)0, qqd,  false, false);
  saed = __builtin_amdgcn_wmma_f32_16x16x4_f32(false, aS2, false, bS2, (short)0, saed, false, false);

  // =====================================================================
  // Stage 3a: exclusive lane-prefix (the triu(k=1) pair mask) as a
  // triangular-mask GEMM on the matrix pipe:  S = L . X,
  //   L (32x32) strict lower-triangular ones (built from lane ids, no LDS)
  //   X (32x24): X[b][(n,d)] = phi_b[n] * r_b[d]
  // Output scattered in-place over X (all B loads precede first store).
  // =====================================================================
  float gv[8][3];
#pragma unroll
  for (int m = 0; m < 8; ++m) {
    const float M0 = R_M(m, 0), M1 = R_M(m, 1), M2 = R_M(m, 2);
    const float w0 = mdy * M2 - mdz * M1;
    const float w1 = mdz * M0 - mdx * M2;
    const float w2 = mdx * M1 - mdy * M0;
    gv[m][0] = rr[1] * w2 - rr[2] * w1;
    gv[m][1] = rr[2] * w0 - rr[0] * w2;
    gv[m][2] = rr[0] * w1 - rr[1] * w0;
  }
#pragma unroll
  for (int d = 0; d < 3; ++d)
#pragma unroll
    for (int n = 0; n < 8; ++n) {
      XS[lane * 24 + d * 8 + n] = phi[n] * rr[d];     // X: K row = lane, col = d*8+n
      YS[lane * 24 + d * 8 + n] = phi[n] * gv[n][d];  // Y for stage 3b
    }
  __syncthreads();

  v8f S00 = {}, S01 = {}, S10 = {}, S11 = {};
#pragma unroll
  for (int s = 0; s < 8; ++s) {
    const int kg0 = s * 4 + hi2;
    const int kg1 = kg0 + 1;
    v2f b0, b1, a0, a1;
    b0.x = XS[kg0 * 24 + row16];       b0.y = XS[kg1 * 24 + row16];
    b1.x = XS[kg0 * 24 + 16 + row16];  b1.y = XS[kg1 * 24 + 16 + row16];
    a0.x = (kg0 < row16) ? 1.0f : 0.0f;        // row tile 0: rows 0-15
    a0.y = (kg1 < row16) ? 1.0f : 0.0f;
    a1.x = (kg0 < row16 + 16) ? 1.0f : 0.0f;   // row tile 1: rows 16-31
    a1.y = (kg1 < row16 + 16) ? 1.0f : 0.0f;
    S00 = __builtin_amdgcn_wmma_f32_16x16x4_f32(false, a0, false, b0, (short)0, S00, false, false);
    S01 = __builtin_amdgcn_wmma_f32_16x16x4_f32(false, a0, false, b1, (short)0, S01, false, false);
    S10 = __builtin_amdgcn_wmma_f32_16x16x4_f32(false, a1, false, b0, (short)0, S10, false, false);
    S11 = __builtin_amdgcn_wmma_f32_16x16x4_f32(false, a1, false, b1, (short)0, S11, false, false);
  }
  {
    const int rowhalf = lo ? 0 : 8;    // D rows: lanes 0-15 -> v, lanes 16-31 -> 8+v
#pragma unroll
    for (int v = 0; v < 8; ++v) {
      XS[(rowhalf + v) * 24 + row16]      = S00[v];   // cols 0-15
      XS[(16 + rowhalf + v) * 24 + row16] = S10[v];
    }
    if (row16 < 8) {                   // cols 16-23 (cols 24-31 of tile 1 unused)
#pragma unroll
      for (int v = 0; v < 8; ++v) {
        XS[(rowhalf + v) * 24 + 16 + row16]      = S01[v];
        XS[(16 + rowhalf + v) * 24 + 16 + row16] = S11[v];
      }
    }
  }
  __syncthreads();

  // =====================================================================
  // Stage 3b: DMI pair term as K=96 GEMM (exact K, no padding, no masks).
  //   d_dmi[n,m] = sum_{b,d} S_b[n,d] * (phi_bm * (r_b x (m_dir x M[m]))_d)
  // =====================================================================
  v8f ddm = {};
#pragma unroll
  for (int s = 0; s < 24; ++s) {
    const int k0 = (s * 4 + hi2) * 8 + r8;
    const int k1 = k0 + 8;
    v2f a, b;
    a.x = XS[k0];  a.y = XS[k1];
    b.x = YS[k0];  b.y = YS[k1];
    ddm = __builtin_amdgcn_wmma_f32_16x16x4_f32(false, a, false, b, (short)0, ddm, false, false);
  }

  // =====================================================================
  // Energy accumulation (fused normalize + weights; output = N floats)
  // =====================================================================
  float eacc = 0.0f;

  if (lane < 8) {                     // WMMA D: lane = column m/n, VGPR = row
#pragma unroll
    for (int n = 0; n < 8; ++n) {
      if (n <= lane) {
        const int t = n * 8 - (n * (n + 1)) / 2 + lane;   // triu index
        eacc += nrm(ppd[n], 8 + t, ss, ssc, wst);         // d_ang_l1
        eacc += nrm(qqd[n], 44 + t, ss, ssc, wst);        // d_ang_l2
      }
      eacc += nrm(saed[n], 19 + n * 8 + lane, ms, msc, wmg);        // d_sae[m=n,lane]
      eacc += nrm(ddm[n], 83 + n * 8 + lane, ms, msc, wmg);         // d_dmi
      eacc += nrm(u * ddm[n], 163 + n * 8 + lane, ms, msc, wmg);    // u*d_dmi
    }
  }

  if (lane == 0) {
    float es = bst[0] + bmg[0] + esh[spi];
#pragma unroll
    for (int t = 0; t < 16; ++t) es += semb[spi * 16 + t] * (wst[t] + wmg[t]);
#pragma unroll
    for (int n = 0; n < 8; ++n) es += nrm(R_DR(n), n, ss, ssc, wst);   // d_radial
    es += nrm(u, 0, ms, msc, wmg);                                     // d_amp
    es += nrm(u * u, 1, ms, msc, wmg);
    es += nrm(u * u * u, 2, ms, msc, wmg);
#pragma unroll
    for (int n = 0; n < 8; ++n) {
      const float diso = mdx * R_M(n, 0) + mdy * R_M(n, 1) + mdz * R_M(n, 2);
      const float a0 = R_Q(n, 0), a1 = R_Q(n, 1), a2 = R_Q(n, 2);
      const float a3 = R_Q(n, 3), a4 = R_Q(n, 4), a5 = R_Q(n, 5);
      const float dsia = a0 * mdx * mdx + a1 * mdy * mdy + a2 * mdz * mdz
                       + 2.0f * (a3 * mdx * mdy + a4 * mdx * mdz + a5 * mdy * mdz);
      const float dex = mdx * R_W(n, 0) + mdy * R_W(n, 1) + mdz * R_W(n, 2);
      es += nrm(diso, 3 + n, ms, msc, wmg)  + nrm(u * diso, 147 + n, ms, msc, wmg);
      es += nrm(dsia, 11 + n, ms, msc, wmg) + nrm(u * dsia, 155 + n, ms, msc, wmg);
      es += nrm(R_DNB(n), 227 + n, ms, msc, wmg);
      es += nrm(dex, 235 + n, ms, msc, wmg) + nrm(u * dex, 243 + n, ms, msc, wmg);
    }
    eacc += es;
  }

  const float etot = wred32(eacc);
  if (lane == 0 && atom < N) out[atom] = etot;
}

extern "C" void kernel_launch(void* const* d_in, const int* in_sizes, int n_in,
                              void* d_out, int out_size, void* d_ws, size_t ws_size,
                              hipStream_t stream) {
  const float* pos    = (const float*)d_in[0];
  const int*   spec   = (const int*)d_in[1];
  const float* mg     = (const float*)d_in[2];
  const int*   eidx   = (const int*)d_in[3];
  const float* shifts = (const float*)d_in[4];
  const float* coeff  = (const float*)d_in[5];
  const float* semb   = (const float*)d_in[6];
  const float* ss     = (const float*)d_in[7];
  const float* ssc    = (const float*)d_in[8];
  const float* ms     = (const float*)d_in[9];
  const float* msc    = (const float*)d_in[10];
  const float* esh    = (const float*)d_in[11];
  const float* wst    = (const float*)d_in[12];
  const float* bst    = (const float*)d_in[13];
  const float* wmg    = (const float*)d_in[14];
  const float* bmg    = (const float*)d_in[15];
  const int N = in_sizes[0] / 3;
  dim3 grid((N + 7) / 8), block(256);
  hipLaunchKernelGGL(magpot_fused, grid, block, 0, stream,
                     pos, spec, mg, eidx, shifts, coeff, semb, ss, ssc, ms, msc,
                     esh, wst, bst, wmg, bmg, (float*)d_out, N);
}